// HierFrameNetwork_9663676416054
// MI455X (gfx1250) — compile-verified
//
#include <hip/hip_runtime.h>

#define HID_S   128
#define HID_V   16
#define EDGE_S  64
#define KNN_K   24
#define EPSF    1e-8f
#define BATCH   32
#define NTOT    1024
#define MSG     48          // 2 nodes * 24 messages per block
#define SPAD    228         // LDS row stride for GEMM A (conflict-free, 8B aligned)
#define SPAD2   164

typedef __attribute__((ext_vector_type(2))) float v2f;
typedef __attribute__((ext_vector_type(8))) float v8f;

struct StageParams {
  const float* wh[3];
  const float* wsb[3];
  const float* wswp[3];   // packed ws_w (WMMA B-fragment layout, zero padded)
  const float* wv[3];
};

// ------------------------------------------------------------- weight packer
// Packed layout: for k-group g (4 k's), col in [0,128), kq in {0,1}:
//   wp[((g*128 + col)*2 + kq)*2 + j] = w[(4g + 2kq + j)*128 + col]  (0 if k>=indim)
// A lane (col = ct*16 + (l&15), kq = l>>4) then loads its B fragment as one b64.
__global__ __launch_bounds__(256)
void pack_w_kernel(const float* __restrict__ w, int indim,
                   float* __restrict__ wp) {
  int kpad = (indim + 3) & ~3;
  int total = (kpad >> 2) * 512;                  // groups * 128 * 2 * 2
  for (int e = blockIdx.x * blockDim.x + threadIdx.x; e < total;
       e += gridDim.x * blockDim.x) {
    int j   = e & 1;
    int kq  = (e >> 1) & 1;
    int col = (e >> 2) & 127;
    int g   = e >> 9;
    int k   = 4*g + 2*kq + j;
    wp[e] = (k < indim) ? w[k*HID_S + col] : 0.f;
  }
}

// ---------------------------------------------------------------- kNN top-24
__global__ __launch_bounds__(256)
void knn_kernel(const float* __restrict__ p, int N1, int N0,
                int* __restrict__ idx_out) {
  int g = blockIdx.x * blockDim.x + threadIdx.x;
  if (g >= BATCH * N1) return;
  int b = g / N1, i = g % N1;
  const float* pb = p + (size_t)b * NTOT * 3;
  float px = pb[i*3+0], py = pb[i*3+1], pz = pb[i*3+2];
  float bd[KNN_K]; int bi[KNN_K];
#pragma unroll
  for (int k = 0; k < KNN_K; ++k) { bd[k] = 3.4e38f; bi[k] = 0; }
  for (int j = 0; j < N0; ++j) {
    float dx = pb[j*3+0]-px, dy = pb[j*3+1]-py, dz = pb[j*3+2]-pz;
    float d2 = dx*dx + dy*dy + dz*dz;
    if (d2 < bd[KNN_K-1]) {
      int k = KNN_K-1;
      while (k > 0 && bd[k-1] > d2) { bd[k]=bd[k-1]; bi[k]=bi[k-1]; --k; }
      bd[k] = d2; bi[k] = j;
    }
  }
  int* o = idx_out + (size_t)g * KNN_K;
#pragma unroll
  for (int k = 0; k < KNN_K; ++k) o[k] = bi[k];
}

// ------------------------------------------------- fused 3-layer GVP message conv
__global__ __launch_bounds__(256)
void gvp_conv_kernel(const float* __restrict__ p, int N1, int first,
                     const int* __restrict__ knn,
                     const float* __restrict__ s_in, const float* __restrict__ v_in,
                     float* __restrict__ s_out, float* __restrict__ v_out,
                     StageParams sp) {
  extern __shared__ float smem[];
  float* sIn   = smem;                  // MSG*SPAD  : GEMM A (concat[s, vn])
  float* sO    = sIn  + MSG*SPAD;       // MSG*HID_S : GEMM out
  float* vCur  = sO   + MSG*HID_S;      // MSG*51    : current v (<=17 rows x 3)
  float* vH    = vCur + MSG*51;         // MSG*51    : vh
  float* dvec  = vH   + MSG*51;         // MSG*3
  float* ddist = dvec + MSG*3;          // MSG

  const int tid   = threadIdx.x;
  const int gnode = blockIdx.x * 2;     // flattened over B*N1, N1 even

  // geometry per message
  if (tid < MSG) {
    int ln = tid / KNN_K, k = tid % KNN_K;
    int gi = gnode + ln;
    int b = gi / N1, i = gi % N1;
    int nbr = knn[gi * KNN_K + k];
    const float* pb = p + (size_t)b * NTOT * 3;
    float dx = pb[nbr*3+0] - pb[i*3+0];
    float dy = pb[nbr*3+1] - pb[i*3+1];
    float dz = pb[nbr*3+2] - pb[i*3+2];
    dvec[tid*3+0] = dx; dvec[tid*3+1] = dy; dvec[tid*3+2] = dz;
    ddist[tid] = sqrtf(dx*dx + dy*dy + dz*dz + EPSF);
  }
  __syncthreads();

  // m_s = [gather(s), rbf(dist)]
  for (int e = tid; e < MSG*(HID_S+EDGE_S); e += 256) {
    int m = e / (HID_S+EDGE_S), c = e % (HID_S+EDGE_S);
    float val;
    if (c < HID_S) {
      if (first) val = 0.f;
      else {
        int gi = gnode + m / KNN_K;
        int b = gi / N1;
        int nbr = knn[gi * KNN_K + (m % KNN_K)];
        val = s_in[((size_t)b*NTOT + nbr)*HID_S + c];
      }
    } else {
      int r = c - HID_S;
      float d = ddist[m] * 10.f;
      if (r == 0)              val = (d <  0.f)  ? 1.f : 0.f;
      else if (r == EDGE_S-1)  val = (d >= 20.f) ? 1.f : 0.f;
      else {
        const float spg = 20.f / 61.f;
        float t = d - spg * (float)(r - 1);
        val = expf((-0.5f/(spg*spg)) * t * t);
      }
    }
    sIn[m*SPAD + c] = val;
  }
  // m_v = [gather(v), unit(d_ij)]
  for (int e = tid; e < MSG*17*3; e += 256) {
    int m = e / 51, r = e % 51;
    int hh = r / 3, c = r % 3;
    float val;
    if (hh < HID_V) {
      if (first) val = 0.f;
      else {
        int gi = gnode + m / KNN_K;
        int b = gi / N1;
        int nbr = knn[gi * KNN_K + (m % KNN_K)];
        val = v_in[(((size_t)b*NTOT + nbr)*HID_V + hh)*3 + c];
      }
    } else {
      val = dvec[m*3+c] / ddist[m];
    }
    vCur[m*51 + hh*3 + c] = val;
  }
  __syncthreads();

  const int lane = tid & 31, wave = tid >> 5;
  const int kq   = lane >> 4;
  const int col  = wave*16 + (lane & 15);

  for (int li = 0; li < 3; ++li) {
    const int si    = (li == 0) ? (HID_S + EDGE_S) : HID_S;
    const int vi    = (li == 0) ? (HID_V + 1) : HID_V;
    const int h     = vi;                 // h = max(vi, 16)
    const int indim = si + h;             // 209 or 144
    const int kpad  = (indim + 3) & ~3;
    const bool act  = (li < 2);
    const float* wh  = sp.wh[li];
    const float* wsb = sp.wsb[li];
    const float* wvp = sp.wv[li];

    // vh[m][h][c] = sum_v v[m][v][c] * wh[v][h]
    for (int e = tid; e < MSG*h*3; e += 256) {
      int m = e / (h*3), r = e % (h*3);
      int hh = r / 3, c = r % 3;
      float acc = 0.f;
      for (int vv = 0; vv < vi; ++vv)
        acc += vCur[m*51 + vv*3 + c] * wh[vv*h + hh];
      vH[m*51 + hh*3 + c] = acc;
    }
    __syncthreads();
    // vn into sIn[:, si..si+h) and zero-pad to kpad
    {
      int w = kpad - si;
      for (int e = tid; e < MSG*w; e += 256) {
        int m = e / w, hh = e % w;
        float val = 0.f;
        if (hh < h) {
          float x = vH[m*51+hh*3+0], y = vH[m*51+hh*3+1], z = vH[m*51+hh*3+2];
          val = sqrtf(x*x + y*y + z*z + EPSF);
        }
        sIn[m*SPAD + si + hh] = val;
      }
    }
    __syncthreads();

    // s GEMM: (48 x kpad) @ (kpad x 128) with V_WMMA_F32_16X16X4_F32
    // A: one ds b64 per step; B: one global b64 per step (pre-packed).
    {
      const float bias = wsb[col];
      const int   ng   = kpad >> 2;
      const float* bbase = sp.wswp[li] + ((size_t)((col << 1) + kq) << 1);
      for (int rt = 0; rt < 3; ++rt) {
        v8f acc = {0.f,0.f,0.f,0.f,0.f,0.f,0.f,0.f};
        const float* arow = sIn + (rt*16 + (lane & 15))*SPAD + 2*kq;
        for (int g = 0; g < ng; ++g) {
          v2f a  = *(const v2f*)(arow + 4*g);
          v2f bf = *(const v2f*)(bbase + (size_t)g*512);
          acc = __builtin_amdgcn_wmma_f32_16x16x4_f32(false, a, false, bf,
                                                      (short)0, acc, false, false);
        }
        int mb = rt*16 + ((lane & 16) ? 8 : 0);
#pragma unroll
        for (int r = 0; r < 8; ++r) {
          float x = acc[r] + bias;
          if (act) x = fmaxf(x, 0.f);
          sO[(mb + r)*HID_S + col] = x;
        }
      }
    }
    __syncthreads();

    // v_out[m][v][c] = sum_h vh[m][h][c] * wv[h][v]
    for (int e = tid; e < MSG*HID_V*3; e += 256) {
      int m = e / (HID_V*3), r = e % (HID_V*3);
      int vv = r / 3, c = r % 3;
      float acc = 0.f;
      for (int hh = 0; hh < h; ++hh)
        acc += vH[m*51 + hh*3 + c] * wvp[hh*HID_V + vv];
      vCur[m*51 + vv*3 + c] = acc;
    }
    __syncthreads();
    if (act) {   // v *= sigmoid(||v||_eps)
      for (int e = tid; e < MSG*HID_V; e += 256) {
        int m = e / HID_V, vv = e % HID_V;
        float x = vCur[m*51+vv*3+0], y = vCur[m*51+vv*3+1], z = vCur[m*51+vv*3+2];
        float n = sqrtf(x*x + y*y + z*z + EPSF);
        float sg = 1.f / (1.f + expf(-n));
        vCur[m*51+vv*3+0] = x*sg;
        vCur[m*51+vv*3+1] = y*sg;
        vCur[m*51+vv*3+2] = z*sg;
      }
      __syncthreads();
    }
    if (li < 2) {  // next layer s input
      for (int e = tid; e < MSG*HID_S; e += 256) {
        int m = e / HID_S, c = e % HID_S;
        sIn[m*SPAD + c] = sO[m*HID_S + c];
      }
      __syncthreads();
    }
  }

  // mean over K=24, write node features
  {
    int ln = tid / HID_S, c = tid % HID_S;   // 256 threads = 2 nodes * 128
    float acc = 0.f;
    for (int k = 0; k < KNN_K; ++k) acc += sO[(ln*KNN_K + k)*HID_S + c];
    int gi = gnode + ln;
    int b = gi / N1, i = gi % N1;
    s_out[((size_t)b*NTOT + i)*HID_S + c] = acc * (1.f/24.f);
  }
  if (tid < 2*HID_V*3) {
    int ln = tid / (HID_V*3), r = tid % (HID_V*3);
    int vv = r / 3, c = r % 3;
    float acc = 0.f;
    for (int k = 0; k < KNN_K; ++k) acc += vCur[(ln*KNN_K + k)*51 + vv*3 + c];
    int gi = gnode + ln;
    int b = gi / N1, i = gi % N1;
    v_out[(((size_t)b*NTOT + i)*HID_V + vv)*3 + c] = acc * (1.f/24.f);
  }
}

// ---------------------------------------------- final GVP + orthonormal frame
__global__ __launch_bounds__(256)
void gvp_out_kernel(const float* __restrict__ s_feat, const float* __restrict__ v_feat,
                    const float* __restrict__ wh, const float* __restrict__ wsb,
                    const float* __restrict__ wswp, const float* __restrict__ wv,
                    float* __restrict__ out) {
  __shared__ float sIn[32*SPAD2];
  __shared__ float sO [32*HID_S];
  __shared__ float vH [32*48];
  const int tid  = threadIdx.x;
  const int base = blockIdx.x * 32;

  for (int e = tid; e < 32*48; e += 256) {
    int ln = e / 48, r = e % 48;
    int hh = r / 3, c = r % 3;
    float acc = 0.f;
    size_t nb = (size_t)(base + ln) * HID_V;
    for (int vv = 0; vv < HID_V; ++vv)
      acc += v_feat[(nb + vv)*3 + c] * wh[vv*16 + hh];
    vH[ln*48 + hh*3 + c] = acc;
  }
  for (int e = tid; e < 32*HID_S; e += 256) {
    int ln = e / HID_S, c = e % HID_S;
    sIn[ln*SPAD2 + c] = s_feat[(size_t)(base + ln)*HID_S + c];
  }
  __syncthreads();
  for (int e = tid; e < 32*16; e += 256) {
    int ln = e / 16, hh = e % 16;
    float x = vH[ln*48+hh*3+0], y = vH[ln*48+hh*3+1], z = vH[ln*48+hh*3+2];
    sIn[ln*SPAD2 + HID_S + hh] = sqrtf(x*x + y*y + z*z + EPSF);
  }
  __syncthreads();

  const int lane = tid & 31, wave = tid >> 5;
  const int kq   = lane >> 4;
  const int col  = wave*16 + (lane & 15);
  const float bias = wsb[col];
  const float* bbase = wswp + ((size_t)((col << 1) + kq) << 1);
  for (int rt = 0; rt < 2; ++rt) {
    v8f acc = {0.f,0.f,0.f,0.f,0.f,0.f,0.f,0.f};
    const float* arow = sIn + (rt*16 + (lane & 15))*SPAD2 + 2*kq;
    for (int g = 0; g < 36; ++g) {              // kpad = 144
      v2f a  = *(const v2f*)(arow + 4*g);
      v2f bf = *(const v2f*)(bbase + (size_t)g*512);
      acc = __builtin_amdgcn_wmma_f32_16x16x4_f32(false, a, false, bf,
                                                  (short)0, acc, false, false);
    }
    int mb = rt*16 + ((lane & 16) ? 8 : 0);
#pragma unroll
    for (int r = 0; r < 8; ++r)
      sO[(mb + r)*HID_S + col] = acc[r] + bias;
  }
  __syncthreads();

  const size_t RN = (size_t)BATCH * NTOT * 9;
  if (tid < 32) {
    int ln = tid;
    float vo[2][3];
    for (int j = 0; j < 2; ++j)
      for (int c = 0; c < 3; ++c) {
        float acc = 0.f;
        for (int hh = 0; hh < 16; ++hh)
          acc += vH[ln*48 + hh*3 + c] * wv[hh*2 + j];
        vo[j][c] = acc;
      }
    float n1 = sqrtf(vo[0][0]*vo[0][0] + vo[0][1]*vo[0][1] + vo[0][2]*vo[0][2] + EPSF);
    float e1[3] = {vo[0][0]/n1, vo[0][1]/n1, vo[0][2]/n1};
    float dp = e1[0]*vo[1][0] + e1[1]*vo[1][1] + e1[2]*vo[1][2];
    float u2[3] = {vo[1][0]-dp*e1[0], vo[1][1]-dp*e1[1], vo[1][2]-dp*e1[2]};
    float n2 = sqrtf(u2[0]*u2[0] + u2[1]*u2[1] + u2[2]*u2[2] + EPSF);
    float e2[3] = {u2[0]/n2, u2[1]/n2, u2[2]/n2};
    float e3[3] = {e1[1]*e2[2]-e1[2]*e2[1],
                   e1[2]*e2[0]-e1[0]*e2[2],
                   e1[0]*e2[1]-e1[1]*e2[0]};
    float* R = out + (size_t)(base + ln) * 9;
    for (int c = 0; c < 3; ++c) { R[c*3+0]=e1[c]; R[c*3+1]=e2[c]; R[c*3+2]=e3[c]; }
  }
  for (int e = tid; e < 32*HID_S; e += 256) {
    int ln = e / HID_S, c = e % HID_S;
    out[RN + (size_t)(base + ln)*HID_S + c] = sO[ln*HID_S + c];
  }
}

// ------------------------------------------------------------------- launch
static void pick4(void* const* d_in, const int* in_sizes, int q,
                  const float** wh, const float** wsb,
                  const float** wsw, const float** wv) {
  // per-layer group: wh first, wv last (both key orders agree);
  // middle two disambiguated by size (ws_b has 128 elements).
  *wh = (const float*)d_in[q];
  *wv = (const float*)d_in[q+3];
  if (in_sizes[q+1] == 128) {
    *wsb = (const float*)d_in[q+1]; *wsw = (const float*)d_in[q+2];
  } else {
    *wsw = (const float*)d_in[q+1]; *wsb = (const float*)d_in[q+2];
  }
}

extern "C" void kernel_launch(void* const* d_in, const int* in_sizes, int n_in,
                              void* d_out, int out_size, void* d_ws, size_t ws_size,
                              hipStream_t stream) {
  (void)n_in; (void)out_size; (void)ws_size;
  const float* p = (const float*)d_in[0];

  const float *wh_[13], *wsb_[13], *wsw_[13], *wv_[13];  // 12 conv + out
  int q = 1;
  for (int i = 0; i < 13; ++i) {
    pick4(d_in, in_sizes, q, &wh_[i], &wsb_[i], &wsw_[i], &wv_[i]);
    q += 4;
  }

  float* ws = (float*)d_ws;
  const size_t SFEAT = (size_t)BATCH*NTOT*HID_S;
  const size_t VFEAT = (size_t)BATCH*NTOT*HID_V*3;
  float* sbuf[2] = { ws, ws + SFEAT };
  float* vbuf[2] = { ws + 2*SFEAT, ws + 2*SFEAT + VFEAT };
  int*   knnbuf  = (int*)(ws + 2*SFEAT + 2*VFEAT);
  float* packb   = ws + 2*SFEAT + 2*VFEAT + (size_t)BATCH*NTOT*KNN_K;

  // pack all 13 ws_w into WMMA B-fragment layout
  StageParams sp[4];
  const float* wswp_[13];
  size_t po = 0;
  for (int i = 0; i < 13; ++i) {
    int indim = (i < 12 && (i % 3) == 0) ? 209 : 144;   // conv layer0: 192+17
    int total = (((indim + 3) & ~3) >> 2) * 512;
    pack_w_kernel<<<(total + 255)/256, 256, 0, stream>>>(wsw_[i], indim, packb + po);
    wswp_[i] = packb + po;
    po += (size_t)total;
  }
  for (int s = 0; s < 4; ++s)
    for (int l = 0; l < 3; ++l) {
      sp[s].wh[l]   = wh_[s*3 + l];
      sp[s].wsb[l]  = wsb_[s*3 + l];
      sp[s].wswp[l] = wswp_[s*3 + l];
      sp[s].wv[l]   = wv_[s*3 + l];
    }

  const int HIERv[5] = {1024, 256, 64, 256, 1024};
  const size_t lds = (size_t)(MSG*SPAD + MSG*HID_S + 2*MSG*51 + MSG*3 + MSG) * sizeof(float);

  for (int st = 0; st < 4; ++st) {
    int N0 = HIERv[st], N1 = HIERv[st+1];
    int nn = BATCH * N1;
    knn_kernel<<<(nn + 255)/256, 256, 0, stream>>>(p, N1, N0, knnbuf);
    gvp_conv_kernel<<<nn/2, 256, lds, stream>>>(p, N1, (st == 0) ? 1 : 0, knnbuf,
        sbuf[st & 1], vbuf[st & 1], sbuf[(st + 1) & 1], vbuf[(st + 1) & 1], sp[st]);
  }
  gvp_out_kernel<<<BATCH*NTOT/32, 256, 0, stream>>>(sbuf[0], vbuf[0],
      wh_[12], wsb_[12], wswp_[12], wv_[12], (float*)d_out);
}